// BiAffineAttention_31473520345597
// MI455X (gfx1250) — compile-verified
//
#include <hip/hip_runtime.h>
#include <hip/hip_bf16.h>

// ---------------------------------------------------------------------------
// BiAffine attention head for MI455X (gfx1250), fp32 WMMA path.
//   B=64, L=512, HID2=800, D_ARC=500, D_REL=100, N_REL=46
// Heavy GEMMs: V_WMMA_F32_16X16X4_F32 (wave32) with async-to-LDS staging of
// the shared B panel (ASYNCcnt / s_wait_asynccnt, double-buffered).
// ---------------------------------------------------------------------------

typedef __attribute__((ext_vector_type(2))) float v2f;
typedef __attribute__((ext_vector_type(8))) float v8f;

__device__ __forceinline__ v8f wmma_f32x4(v2f a, v2f b, v8f c) {
  // 8 args: (neg_a, A, neg_b, B, c_mod, C, reuse_a, reuse_b)
  return __builtin_amdgcn_wmma_f32_16x16x4_f32(
      false, a, false, b, (short)0, c, false, false);
}

// gfx1250 async global->LDS copy (tracked by ASYNCcnt).
__device__ __forceinline__ void async_b128(unsigned lds_off, const float* g) {
  asm volatile("global_load_async_to_lds_b128 %0, %1, off"
               :: "v"(lds_off), "v"(g)
               : "memory");
}
__device__ __forceinline__ void wait_async_le2() {
  asm volatile("s_wait_asynccnt 0x2" ::: "memory");
}
__device__ __forceinline__ void wait_async_0() {
  asm volatile("s_wait_asynccnt 0x0" ::: "memory");
}

// LDS staging geometry (floats). BT panel: [64 cols][pitch 20] (16B-aligned
// rows, 16-lane column reads hit 16 distinct banks). BN panel: [16 k][pitch 68].
#define BT_PITCH 20
#define BN_PITCH 68
#define BUF_FLTS 1280  // max(64*20, 16*68); per buffer
// Stage one 16(K) x 64(N) B chunk into LDS, whole 128-thread block.
// BT=true : B is [N,K] row-major; store Bs[col][k]  (contig k per col)
// BT=false: B is [K,N] row-major; store Bs[k][col]  (contig col per k)
template <bool BT>
__device__ __forceinline__ void stage_B(const float* __restrict__ Bm, int ldb,
                                        int n0, int kc, unsigned bufbase,
                                        int tid) {
  if (BT) {
    const int col = tid >> 1, half = tid & 1;       // 2 threads per column
    const float* g = Bm + (long long)(n0 + col) * ldb + kc + half * 8;
    const unsigned l = bufbase + (unsigned)(col * BT_PITCH + half * 8) * 4u;
    async_b128(l, g);
    async_b128(l + 16u, g + 4);
  } else {
    const int row = tid >> 3, seg = tid & 7;        // 8 threads per k-row
    const float* g = Bm + (long long)(kc + row) * ldb + n0 + seg * 8;
    const unsigned l = bufbase + (unsigned)(row * BN_PITCH + seg * 8) * 4u;
    async_b128(l, g);
    async_b128(l + 16u, g + 4);
  }
}

// ---------------------------------------------------------------------------
// Batched WMMA GEMM: C[z][m][n] = act( A[z] @ B' + bias[z][n] )
//   BT=true : B is [N,K] row-major -> C = A @ B^T
//   BT=false: B is [K,N] row-major -> C = A @ B
// Block = 128 threads (4 waves). Wave w owns rows [by*128 + w*32, +32),
// cols [bx*64, +64): 8 accumulators (2 row-tiles x 4 col-tiles).
// Requirements: M % 128 == 0, K % 4 == 0. N arbitrary (edge blocks guarded).
// ---------------------------------------------------------------------------
template <bool BT, bool RELU, bool BIAS>
__global__ void ba_gemm_wmma(const float* __restrict__ A, int lda, long long sA,
                             const float* __restrict__ Bm, int ldb, long long sB,
                             const float* __restrict__ bias, long long sBias,
                             float* __restrict__ C, int ldc, long long sC,
                             int M, int N, int K) {
  __shared__ float Bs[2 * BUF_FLTS];

  const int z = blockIdx.z;
  A  += (long long)z * sA;
  Bm += (long long)z * sB;
  C  += (long long)z * sC;
  const float* biasz = BIAS ? (bias + (long long)z * sBias) : nullptr;

  const int tid  = threadIdx.x;
  const int lane = tid & 31;
  const int wave = tid >> 5;
  const int h  = lane >> 4;   // half-wave
  const int ln = lane & 15;

  const int m0 = blockIdx.y * 128 + wave * 32;
  const int n0 = blockIdx.x * 64;

  const v8f vzero = {};
  v8f acc0[4], acc1[4];
#pragma unroll
  for (int t = 0; t < 4; ++t) { acc0[t] = vzero; acc1[t] = vzero; }

  const float* A0row = A + (long long)(m0 + ln) * lda;        // rows m0..m0+15
  const float* A1row = A + (long long)(m0 + 16 + ln) * lda;   // rows m0+16..+31

  const bool full = (n0 + 64 <= N);

  if (full) {
    // ---- async-to-LDS double-buffered main loop over 16-wide K chunks ----
    const unsigned ldsB = (unsigned)(uintptr_t)Bs;
    const int nch = K / 16;
    if (nch > 0) stage_B<BT>(Bm, ldb, n0, 0, ldsB, tid);
    for (int c = 0; c < nch; ++c) {
      if (c + 1 < nch) {
        stage_B<BT>(Bm, ldb, n0, (c + 1) * 16,
                    ldsB + (unsigned)(((c + 1) & 1) * BUF_FLTS * 4), tid);
        wait_async_le2();   // current chunk done; next (2 asyncs) in flight
      } else {
        wait_async_0();
      }
      __syncthreads();
      const float* bsp = Bs + (c & 1) * BUF_FLTS;
#pragma unroll
      for (int ks = 0; ks < 16; ks += 4) {
        const int kg = c * 16 + ks + 2 * h;
        v2f a0, a1;
        a0.x = A0row[kg]; a0.y = A0row[kg + 1];
        a1.x = A1row[kg]; a1.y = A1row[kg + 1];
        const int kl = ks + 2 * h;
#pragma unroll
        for (int t = 0; t < 4; ++t) {
          const int n = t * 16 + ln;
          v2f b;
          if (BT) {
            b.x = bsp[n * BT_PITCH + kl];
            b.y = bsp[n * BT_PITCH + kl + 1];
          } else {
            b.x = bsp[kl * BN_PITCH + n];
            b.y = bsp[(kl + 1) * BN_PITCH + n];
          }
          acc0[t] = wmma_f32x4(a0, b, acc0[t]);
          acc1[t] = wmma_f32x4(a1, b, acc1[t]);
        }
      }
      __syncthreads();
    }
    // ---- K remainder (K % 16), direct global loads (cols in bounds) ----
    for (int k = nch * 16; k < K; k += 4) {
      const int kg = k + 2 * h;
      v2f a0, a1;
      a0.x = A0row[kg]; a0.y = A0row[kg + 1];
      a1.x = A1row[kg]; a1.y = A1row[kg + 1];
#pragma unroll
      for (int t = 0; t < 4; ++t) {
        const int col = n0 + t * 16 + ln;
        v2f b;
        if (BT) {
          const float* bp = Bm + (long long)col * ldb + kg;
          b.x = bp[0]; b.y = bp[1];
        } else {
          b.x = Bm[(long long)kg * ldb + col];
          b.y = Bm[(long long)(kg + 1) * ldb + col];
        }
        acc0[t] = wmma_f32x4(a0, b, acc0[t]);
        acc1[t] = wmma_f32x4(a1, b, acc1[t]);
      }
    }
  } else {
    // ---- partial-N edge block: guarded direct-global path ----
    for (int k = 0; k < K; k += 4) {
      const int kg = k + 2 * h;
      v2f a0, a1;
      a0.x = A0row[kg]; a0.y = A0row[kg + 1];
      a1.x = A1row[kg]; a1.y = A1row[kg + 1];
#pragma unroll
      for (int t = 0; t < 4; ++t) {
        const int col = n0 + t * 16 + ln;
        v2f b;
        if (col < N) {
          if (BT) {
            const float* bp = Bm + (long long)col * ldb + kg;
            b.x = bp[0]; b.y = bp[1];
          } else {
            b.x = Bm[(long long)kg * ldb + col];
            b.y = Bm[(long long)(kg + 1) * ldb + col];
          }
        } else {
          b.x = 0.0f; b.y = 0.0f;
        }
        acc0[t] = wmma_f32x4(a0, b, acc0[t]);
        acc1[t] = wmma_f32x4(a1, b, acc1[t]);
      }
    }
  }

  // ---- Epilogue: acc[t][v] is element (rowtile + v + 8h, n0 + t*16 + ln) ----
#pragma unroll
  for (int t = 0; t < 4; ++t) {
    const int col = n0 + t * 16 + ln;
    if (col >= N) continue;
    const float bv = BIAS ? biasz[col] : 0.0f;
#pragma unroll
    for (int v = 0; v < 8; ++v) {
      const int row0 = m0 + v + 8 * h;
      float val0 = acc0[t][v] + bv;
      float val1 = acc1[t][v] + bv;
      if (RELU) { val0 = fmaxf(val0, 0.0f); val1 = fmaxf(val1, 0.0f); }
      C[(long long)row0 * ldc + col] = val0;
      C[(long long)(row0 + 16) * ldc + col] = val1;
    }
  }
}

// ---------------------------------------------------------------------------
// bias[t] = dot(H_arc_head[t,:], b_arc)
// ---------------------------------------------------------------------------
__global__ void ba_head_bias(const float* __restrict__ Hhead,
                             const float* __restrict__ b_arc,
                             float* __restrict__ biasOut, int T, int D) {
  const int t = blockIdx.x * blockDim.x + threadIdx.x;
  if (t >= T) return;
  const float* p = Hhead + (long long)t * D;
  float s = 0.0f;
  for (int d = 0; d < D; ++d) s += p[d] * b_arc[d];
  biasOut[t] = s;
}

// ---------------------------------------------------------------------------
// Row-wise argmax (first max wins, matching jnp.argmax). One wave per row.
// ---------------------------------------------------------------------------
__global__ void ba_argmax_rows(const float* __restrict__ S,
                               float* __restrict__ outF,
                               int* __restrict__ outI, int rows, int cols) {
  const int wave = threadIdx.x >> 5;
  const int lane = threadIdx.x & 31;
  const int row = blockIdx.x * 4 + wave;
  if (row >= rows) return;
  const float* p = S + (long long)row * cols;
  float best = -__builtin_inff();
  int bidx = 0x7fffffff;
  for (int j = lane; j < cols; j += 32) {
    const float v = p[j];
    if (v > best) { best = v; bidx = j; }
  }
#pragma unroll
  for (int m = 16; m >= 1; m >>= 1) {
    const float ov = __shfl_xor(best, m, 32);
    const int oi = __shfl_xor(bidx, m, 32);
    if (ov > best || (ov == best && oi < bidx)) { best = ov; bidx = oi; }
  }
  if (lane == 0) {
    outF[row] = (float)bidx;
    outI[row] = bidx;
  }
}

// ---------------------------------------------------------------------------
// Hg[b,l,:] = H_rel_head[b, head_preds[b,l], :]
// ---------------------------------------------------------------------------
__global__ void ba_gather_hg(const float* __restrict__ RelHead,
                             const int* __restrict__ hp,
                             float* __restrict__ Hg, int Bn, int L, int D) {
  const int idx = blockIdx.x * blockDim.x + threadIdx.x;
  const int total = Bn * L * D;
  if (idx >= total) return;
  const int d = idx % D;
  const int t = idx / D;
  const int b = t / L;
  const int src = b * L + hp[t];
  Hg[idx] = RelHead[(long long)src * D + d];
}

// ---------------------------------------------------------------------------
// S_rel[t,r] = b_rel[r] + sum_d (Hg[t,d] + Hdep[t,d]) * W_rel[d,r]
// ---------------------------------------------------------------------------
__global__ void ba_rel_sums(const float* __restrict__ Hg,
                            const float* __restrict__ Hdep,
                            const float* __restrict__ W_rel,
                            const float* __restrict__ b_rel,
                            float* __restrict__ S_rel, int T, int D, int R) {
  const int idx = blockIdx.x * blockDim.x + threadIdx.x;
  if (idx >= T * R) return;
  const int r = idx % R;
  const int t = idx / R;
  float s = b_rel[r];
  const float* g = Hg + (long long)t * D;
  const float* dd = Hdep + (long long)t * D;
  for (int d = 0; d < D; ++d) s += (g[d] + dd[d]) * W_rel[d * R + r];
  S_rel[idx] = s;
}

// ---------------------------------------------------------------------------
// S_rel[t,r] += sum_j ( sum_i Hg[t,i] * U[i,r,j] ) * Hdep[t,j]
// grid = (T/64, 46); block = 128 (4 waves), wave = one 16-token tile, one r.
// ---------------------------------------------------------------------------
__global__ void ba_interactions_wmma(const float* __restrict__ Hg,
                                     const float* __restrict__ Hdep,
                                     const float* __restrict__ U,  // [100,46,100]
                                     float* __restrict__ S_rel,    // [T,46]
                                     int T) {
  const int DR = 100, NR = 46, LDU = 46 * 100;
  const int r = blockIdx.y;
  const int wave = threadIdx.x >> 5;
  const int lane = threadIdx.x & 31;
  const int h = lane >> 4;
  const int ln = lane & 15;
  const int tok0 = (blockIdx.x * 4 + wave) * 16;
  if (tok0 >= T) return;

  const v8f vzero = {};
  v8f acc[7];
#pragma unroll
  for (int t = 0; t < 7; ++t) acc[t] = vzero;

  const float* arow = Hg + (long long)(tok0 + ln) * DR + 2 * h;
  const float* Ur = U + r * 100;

  for (int k = 0; k < DR; k += 4) {
    v2f a;
    a.x = arow[k];
    a.y = arow[k + 1];
#pragma unroll
    for (int t = 0; t < 7; ++t) {
      const int j = t * 16 + ln;
      v2f b;
      if (j < DR) {
        b.x = Ur[(long long)(k + 2 * h) * LDU + j];
        b.y = Ur[(long long)(k + 2 * h + 1) * LDU + j];
      } else {
        b.x = 0.0f;
        b.y = 0.0f;
      }
      acc[t] = wmma_f32x4(a, b, acc[t]);
    }
  }

  float rowsum[8];
#pragma unroll
  for (int v = 0; v < 8; ++v) rowsum[v] = 0.0f;
#pragma unroll
  for (int t = 0; t < 7; ++t) {
    const int j = t * 16 + ln;
    if (j < DR) {
#pragma unroll
      for (int v = 0; v < 8; ++v) {
        const int m = v + 8 * h;
        rowsum[v] += acc[t][v] * Hdep[(long long)(tok0 + m) * DR + j];
      }
    }
  }
#pragma unroll
  for (int v = 0; v < 8; ++v) {
    float s = rowsum[v];
    s += __shfl_xor(s, 1, 32);
    s += __shfl_xor(s, 2, 32);
    s += __shfl_xor(s, 4, 32);
    s += __shfl_xor(s, 8, 32);
    if (ln == 0) {
      const int m = v + 8 * h;
      S_rel[(long long)(tok0 + m) * NR + r] += s;  // exclusive owner: safe RMW
    }
  }
}

// ---------------------------------------------------------------------------
// Host-side orchestration
// ---------------------------------------------------------------------------
extern "C" void kernel_launch(void* const* d_in, const int* in_sizes, int n_in,
                              void* d_out, int out_size, void* d_ws, size_t ws_size,
                              hipStream_t stream) {
  (void)in_sizes; (void)n_in; (void)out_size; (void)ws_size;

  constexpr int Bn = 64, L = 512, HID2 = 800;
  constexpr int D_ARC = 500, D_REL = 100, N_REL = 46;
  constexpr int T = Bn * L;  // 32768 tokens

  const float* H      = (const float*)d_in[0];
  // d_in[1] = sent_lens (unused by the reference math)
  const float* W_ah   = (const float*)d_in[2];
  const float* b_ah   = (const float*)d_in[3];
  const float* W_ad   = (const float*)d_in[4];
  const float* b_ad   = (const float*)d_in[5];
  const float* W_rh   = (const float*)d_in[6];
  const float* b_rh   = (const float*)d_in[7];
  const float* W_rd   = (const float*)d_in[8];
  const float* b_rd   = (const float*)d_in[9];
  const float* W_arc  = (const float*)d_in[10];
  const float* b_arc  = (const float*)d_in[11];
  const float* U_rel  = (const float*)d_in[12];
  const float* W_rel  = (const float*)d_in[13];
  const float* b_rel  = (const float*)d_in[14];

  // Output layout: S_arc [64,512,512] | S_rel [64,512,46] | head_preds [64,512]
  float* out = (float*)d_out;
  float* out_Sarc = out;
  float* out_Srel = out + (long long)T * L;
  float* out_hpF  = out_Srel + (long long)T * N_REL;

  // Workspace carve-up (floats; all region starts 16B-aligned)
  float* ws = (float*)d_ws;
  float* arc_head = ws;                                 // 32768*500
  float* arc_dep  = arc_head + (long long)T * D_ARC;
  float* rel_head = arc_dep  + (long long)T * D_ARC;    // 32768*100
  float* rel_dep  = rel_head + (long long)T * D_REL;
  float* Tbuf     = rel_dep  + (long long)T * D_REL;    // 32768*500
  float* biasbuf  = Tbuf     + (long long)T * D_ARC;    // 32768
  float* Hg       = biasbuf  + T;                       // 32768*100
  int*   hp_int   = (int*)(Hg + (long long)T * D_REL);  // 32768

  const dim3 blk(128, 1, 1);

  // ---- 1) Four fused projection GEMMs with ReLU: X = relu(H @ W^T + b) ----
  {
    dim3 gArc((D_ARC + 63) / 64, T / 128, 1);   // (8, 256)
    ba_gemm_wmma<true, true, true><<<gArc, blk, 0, stream>>>(
        H, HID2, 0, W_ah, HID2, 0, b_ah, 0, arc_head, D_ARC, 0, T, D_ARC, HID2);
    ba_gemm_wmma<true, true, true><<<gArc, blk, 0, stream>>>(
        H, HID2, 0, W_ad, HID2, 0, b_ad, 0, arc_dep, D_ARC, 0, T, D_ARC, HID2);
    dim3 gRel((D_REL + 63) / 64, T / 128, 1);   // (2, 256)
    ba_gemm_wmma<true, true, true><<<gRel, blk, 0, stream>>>(
        H, HID2, 0, W_rh, HID2, 0, b_rh, 0, rel_head, D_REL, 0, T, D_REL, HID2);
    ba_gemm_wmma<true, true, true><<<gRel, blk, 0, stream>>>(
        H, HID2, 0, W_rd, HID2, 0, b_rd, 0, rel_dep, D_REL, 0, T, D_REL, HID2);
  }

  // ---- 2) bias[b,j] = H_arc_head[b,j,:] . b_arc ----
  ba_head_bias<<<dim3((T + 255) / 256), dim3(256), 0, stream>>>(
      arc_head, b_arc, biasbuf, T, D_ARC);

  // ---- 3) Tbuf = H_arc_dep @ W_arc  (B in [K,N] layout) ----
  {
    dim3 g((D_ARC + 63) / 64, T / 128, 1);
    ba_gemm_wmma<false, false, false><<<g, blk, 0, stream>>>(
        arc_dep, D_ARC, 0, W_arc, D_ARC, 0, nullptr, 0, Tbuf, D_ARC, 0,
        T, D_ARC, D_ARC);
  }

  // ---- 4) S_arc[b] = Tbuf[b] @ arc_head[b]^T + bias[b,:] (batched, 64) ----
  {
    dim3 g(L / 64, L / 128, Bn);  // (8, 4, 64)
    ba_gemm_wmma<true, false, true><<<g, blk, 0, stream>>>(
        Tbuf, D_ARC, (long long)L * D_ARC,
        arc_head, D_ARC, (long long)L * D_ARC,
        biasbuf, L,
        out_Sarc, L, (long long)L * L,
        L, L, D_ARC);
  }

  // ---- 5) head_preds = argmax_j S_arc[b,i,j] ----
  ba_argmax_rows<<<dim3((T + 3) / 4), blk, 0, stream>>>(
      out_Sarc, out_hpF, hp_int, T, L);

  // ---- 6) Hg = gather(H_rel_head, head_preds) ----
  {
    const int total = T * D_REL;
    ba_gather_hg<<<dim3((total + 255) / 256), dim3(256), 0, stream>>>(
        rel_head, hp_int, Hg, Bn, L, D_REL);
  }

  // ---- 7) S_rel base = (Hg + rel_dep) @ W_rel + b_rel ----
  {
    const int total = T * N_REL;
    ba_rel_sums<<<dim3((total + 255) / 256), dim3(256), 0, stream>>>(
        Hg, rel_dep, W_rel, b_rel, out_Srel, T, D_REL, N_REL);
  }

  // ---- 8) S_rel += bilinear interactions (WMMA) ----
  {
    dim3 g(T / 64, N_REL, 1);  // (512, 46)
    ba_interactions_wmma<<<g, blk, 0, stream>>>(
        Hg, rel_dep, U_rel, out_Srel, T);
  }
}